// ISOC_VGAE_15393162789528
// MI455X (gfx1250) — compile-verified
//
#include <hip/hip_runtime.h>
#include <hip/hip_bf16.h>

typedef __attribute__((ext_vector_type(2))) float v2f;
typedef __attribute__((ext_vector_type(8))) float v8f;

#define TF 128  // feature dim of wide tables
#define NROWS 16

// ---------------------------------------------------------------------------
// CDNA5 async global->LDS copy (per-lane 16B), tracked with ASYNCcnt
// ---------------------------------------------------------------------------
__device__ __forceinline__ void async_copy_b128(unsigned lds_byte_off,
                                                const void* gaddr) {
  asm volatile("global_load_async_to_lds_b128 %0, %1, off"
               :: "v"(lds_byte_off), "v"((unsigned long long)(size_t)gaddr)
               : "memory");
}
__device__ __forceinline__ void wait_async0() {
  asm volatile("s_wait_asynccnt 0x0" ::: "memory");
}
// copy nbytes (multiple of blockDim.x*16) from contiguous global to LDS
__device__ __forceinline__ void async_stage(void* lds_dst, const void* gsrc,
                                            int nbytes) {
  unsigned base = (unsigned)(size_t)lds_dst;  // low 32 bits = LDS offset
  for (int off = threadIdx.x * 16; off < nbytes; off += blockDim.x * 16) {
    async_copy_b128(base + off, (const char*)gsrc + off);
  }
}

// ---------------------------------------------------------------------------
// threefry2x32-style counter RNG + normal transform (deterministic)
// ---------------------------------------------------------------------------
__device__ __forceinline__ unsigned rotl32(unsigned x, int r) {
  return (x << r) | (x >> (32 - r));
}

__device__ __forceinline__ void threefry2x32(unsigned k0, unsigned k1,
                                             unsigned c0, unsigned c1,
                                             unsigned& o0, unsigned& o1) {
  const unsigned rot[8] = {13u, 15u, 26u, 6u, 17u, 29u, 16u, 24u};
  unsigned ks[3] = {k0, k1, k0 ^ k1 ^ 0x1BD11BDAu};
  unsigned x0 = c0 + ks[0];
  unsigned x1 = c1 + ks[1];
#pragma unroll
  for (int blk = 0; blk < 5; ++blk) {
#pragma unroll
    for (int r = 0; r < 4; ++r) {
      x0 += x1;
      x1 = rotl32(x1, rot[(blk & 1) * 4 + r]);
      x1 ^= x0;
    }
    x0 += ks[(blk + 1) % 3];
    x1 += ks[(blk + 2) % 3] + (unsigned)(blk + 1);
  }
  o0 = x0; o1 = x1;
}

__device__ __forceinline__ float noise_normal(unsigned stream, unsigned idx) {
  unsigned r0, r1;
  threefry2x32(42u, stream, idx, 0x9E3779B9u, r0, r1);
  float u1 = ((float)((r0 >> 8) + 1u)) * (1.0f / 16777216.0f);  // (0,1]
  float u2 = ((float)(r1 >> 8)) * (1.0f / 16777216.0f);
  return sqrtf(-2.0f * __logf(u1)) * __cosf(6.28318530718f * u2);
}

// ---------------------------------------------------------------------------
// Fused LDS->LDS MLP layer: OUT[16 x NOUT] = act(A[16 x K] @ W[K x NOUT] + b)
// One wave per 16x16 output tile, K accumulated with V_WMMA_F32_16X16X4_F32.
// ---------------------------------------------------------------------------
template <int K, int NOUT, bool RELU>
__device__ __forceinline__ void mlp_layer(const float* __restrict__ As,
                                          const float* __restrict__ W,
                                          const float* __restrict__ bias,
                                          float* __restrict__ Os) {
  const int lane = threadIdx.x & 31;
  const int wave = threadIdx.x >> 5;
  const int nwaves = blockDim.x >> 5;
  const int m = lane & 15;            // row (A) / col (B,C)
  const int kh = (lane >> 4) << 1;    // 0 for lanes 0-15, 2 for lanes 16-31
  for (int nt = wave * 16; nt < NOUT; nt += nwaves * 16) {
    v8f acc = {0.f, 0.f, 0.f, 0.f, 0.f, 0.f, 0.f, 0.f};
#pragma unroll 8
    for (int k = 0; k < K; k += 4) {
      v2f a, b;
      a[0] = As[m * K + k + kh];
      a[1] = As[m * K + k + kh + 1];
      b[0] = W[(k + kh) * NOUT + nt + m];
      b[1] = W[(k + kh + 1) * NOUT + nt + m];
      acc = __builtin_amdgcn_wmma_f32_16x16x4_f32(false, a, false, b,
                                                  (short)0, acc, false, false);
    }
    float bn = bias[nt + m];
#pragma unroll
    for (int j = 0; j < 8; ++j) {
      int mm = (lane < 16) ? j : (j + 8);
      float v = acc[j] + bn;
      if (RELU) v = fmaxf(v, 0.0f);
      Os[mm * NOUT + nt + m] = v;
    }
  }
}

// ---------------------------------------------------------------------------
// block reduction helper (LDS)
// ---------------------------------------------------------------------------
__device__ __forceinline__ float block_reduce(float v, float* red) {
  red[threadIdx.x] = v;
  __syncthreads();
  for (int s = blockDim.x >> 1; s > 0; s >>= 1) {
    if ((int)threadIdx.x < s) red[threadIdx.x] += red[threadIdx.x + s];
    __syncthreads();
  }
  float r = red[0];
  __syncthreads();
  return r;
}

// ---------------------------------------------------------------------------
// utility kernels
// ---------------------------------------------------------------------------
__global__ void zero_accum_kernel(float* ws) {
  if (threadIdx.x < 16) ws[threadIdx.x] = 0.0f;
}

__global__ void copy_f4_kernel(const float* __restrict__ src,
                               float* __restrict__ dst, long long n4) {
  long long i = (long long)blockIdx.x * blockDim.x + threadIdx.x;
  long long stride = (long long)gridDim.x * blockDim.x;
  for (; i < n4; i += stride) {
    ((float4*)dst)[i] = ((const float4*)src)[i];
  }
}

// scatter-add over edges: acc[dst] += x[src]  (128 features, 4 per thread)
__global__ __launch_bounds__(256) void agg_edge_kernel(
    const float* __restrict__ x, const int* __restrict__ src,
    const int* __restrict__ dst, float* __restrict__ acc, int E) {
  long long idx = (long long)blockIdx.x * blockDim.x + threadIdx.x;
  long long total = (long long)E * 32;
  if (idx >= total) return;
  int e = (int)(idx >> 5);
  int q = (int)(idx & 31);
  int s = src[e], d = dst[e];
  const float4 v = *(const float4*)(x + (long long)s * TF + q * 4);
  float* p = acc + (long long)d * TF + q * 4;
  atomicAdd(p + 0, v.x);
  atomicAdd(p + 1, v.y);
  atomicAdd(p + 2, v.z);
  atomicAdd(p + 3, v.w);
}

// GCN layer: out[16 rows] = act(inv * (acc @ W) + b), WMMA fp32.
// Raw acc tile is async-copied into LDS; row normalization applied in the
// epilogue (linearity: (inv*A)@W == inv*(A@W)).
template <int K, int NOUT, bool RELU, bool DUAL>
__global__ __launch_bounds__(256) void gcn_gemm_kernel(
    const float* __restrict__ A, const int* __restrict__ degree,
    const float* __restrict__ W, const float* __restrict__ bias,
    float* __restrict__ out, float* __restrict__ out2) {
  __shared__ float As[NROWS * K];
  __shared__ float sh_inv[NROWS];
  const int m0 = blockIdx.x * NROWS;
  async_stage(As, A + (long long)m0 * K, NROWS * K * 4);
  if (threadIdx.x < NROWS)
    sh_inv[threadIdx.x] = 1.0f / ((float)degree[m0 + threadIdx.x] + 1.0f);
  wait_async0();
  __syncthreads();

  const int lane = threadIdx.x & 31;
  const int wave = threadIdx.x >> 5;
  const int nwaves = blockDim.x >> 5;
  const int m = lane & 15;
  const int kh = (lane >> 4) << 1;
  for (int nt = wave * 16; nt < NOUT; nt += nwaves * 16) {
    v8f acc = {0.f, 0.f, 0.f, 0.f, 0.f, 0.f, 0.f, 0.f};
#pragma unroll 8
    for (int k = 0; k < K; k += 4) {
      v2f a, b;
      a[0] = As[m * K + k + kh];
      a[1] = As[m * K + k + kh + 1];
      b[0] = W[(k + kh) * NOUT + nt + m];
      b[1] = W[(k + kh + 1) * NOUT + nt + m];
      acc = __builtin_amdgcn_wmma_f32_16x16x4_f32(false, a, false, b,
                                                  (short)0, acc, false, false);
    }
    float bn = bias[nt + m];
#pragma unroll
    for (int j = 0; j < 8; ++j) {
      int mm = (lane < 16) ? j : (j + 8);
      float v = acc[j] * sh_inv[mm] + bn;
      if (RELU) v = fmaxf(v, 0.0f);
      long long o = (long long)(m0 + mm) * NOUT + nt + m;
      out[o] = v;
      if (DUAL) out2[o] = v;
    }
  }
}

// ---------------------------------------------------------------------------
// decoder megakernel 1: rs0 / ds0 / rd0 chains from h2, z1 + loss partials
// accum: [0]=S1 [1]=K1 [2]=D1 [3]=S2 [4]=K2 [5]=D2
// ---------------------------------------------------------------------------
__global__ __launch_bounds__(256) void decoder1_kernel(
    const float* __restrict__ h2w, const float* __restrict__ h1,
    const float* __restrict__ acc1, const int* __restrict__ degree,
    const float* rs0_w1, const float* rs0_b1, const float* rs0_w2, const float* rs0_b2,
    const float* ds0_w1, const float* ds0_b1, const float* ds0_w2, const float* ds0_b2,
    const float* rd0_w1, const float* rd0_b1, const float* rd0_w2, const float* rd0_b2,
    const float* rd0_w3, const float* rd0_b3,
    float* __restrict__ z1, float* __restrict__ accum) {
  __shared__ float sh_h2[NROWS * 64];
  __shared__ float sh_t[NROWS * 64];
  __shared__ float sh_m1[NROWS * 128];
  __shared__ float sh_l1[NROWS * 128];
  __shared__ float sh_rdeg[NROWS];
  __shared__ float sh_inv[NROWS];
  __shared__ float red[256];

  const int m0 = blockIdx.x * NROWS;
  async_stage(sh_h2, h2w + (long long)m0 * 64, NROWS * 64 * 4);
  if (threadIdx.x < NROWS)
    sh_inv[threadIdx.x] = 1.0f / ((float)degree[m0 + threadIdx.x] + 1.0f);
  wait_async0();
  __syncthreads();

  // mean1 = rs0(h2) : 64 -> 64(relu) -> 128
  mlp_layer<64, 64, true>(sh_h2, rs0_w1, rs0_b1, sh_t);   __syncthreads();
  mlp_layer<64, 128, false>(sh_t, rs0_w2, rs0_b2, sh_m1); __syncthreads();
  // log_std1 = ds0(h2)
  mlp_layer<64, 64, true>(sh_h2, ds0_w1, ds0_b1, sh_t);   __syncthreads();
  mlp_layer<64, 128, false>(sh_t, ds0_w2, ds0_b2, sh_l1); __syncthreads();
  // rd0 hidden chain (h2 no longer needed -> reuse its buffer)
  mlp_layer<64, 64, true>(sh_h2, rd0_w1, rd0_b1, sh_t);   __syncthreads();
  mlp_layer<64, 64, true>(sh_t, rd0_w2, rd0_b2, sh_h2);   __syncthreads();
  if (threadIdx.x < NROWS) {
    float s = rd0_b3[0];
    for (int k = 0; k < 64; ++k) s += sh_h2[threadIdx.x * 64 + k] * rd0_w3[k];
    sh_rdeg[threadIdx.x] = fmaxf(s, 0.0f);
  }
  __syncthreads();

  float s1 = 0.0f, k1 = 0.0f;
  for (int i = threadIdx.x; i < NROWS * 128; i += blockDim.x) {
    int m = i >> 7, f = i & 127;
    int node = m0 + m;
    float mean = sh_m1[i], ls = sh_l1[i];
    float nz = noise_normal(0u, (unsigned)(node * 128 + f));
    float zv = mean + nz * __expf(ls);
    z1[(long long)node * 128 + f] = zv;
    float dz = h1[(long long)node * 128 + f] - zv;
    s1 += dz * dz;
    float dm = mean - acc1[(long long)node * 128 + f] * sh_inv[m];
    k1 += -1.0f - 2.0f * ls + dm * dm + __expf(2.0f * ls);
  }
  float S1 = block_reduce(s1, red);
  float K1 = block_reduce(k1, red);
  if (threadIdx.x == 0) {
    atomicAdd(accum + 0, S1);
    atomicAdd(accum + 1, K1);
    float d1 = 0.0f;
    for (int r = 0; r < NROWS; ++r) {
      float dv = sh_rdeg[r] - (float)degree[m0 + r];
      d1 += dv * dv;
    }
    atomicAdd(accum + 2, d1);
  }
}

// ---------------------------------------------------------------------------
// decoder megakernel 2: dm0 / rs1 / ds1 / rd1 chains, z0 + loss partials
// ---------------------------------------------------------------------------
__global__ __launch_bounds__(256) void decoder2_kernel(
    const float* __restrict__ h0, const float* __restrict__ h1,
    const float* __restrict__ z1, const float* __restrict__ acc0,
    const int* __restrict__ degree,
    const float* dm0_w1, const float* dm0_b1, const float* dm0_w2, const float* dm0_b2,
    const float* rs1_w1, const float* rs1_b1, const float* rs1_w2, const float* rs1_b2,
    const float* ds1_w1, const float* ds1_b1, const float* ds1_w2, const float* ds1_b2,
    const float* rd1_w1, const float* rd1_b1, const float* rd1_w2, const float* rd1_b2,
    const float* rd1_w3, const float* rd1_b3,
    float* __restrict__ accum) {
  __shared__ float sh_a[NROWS * 128];   // h1
  __shared__ float sh_b[NROWS * 128];   // z1, later rd1 hidden2
  __shared__ float sh_t[NROWS * 128];   // tmp hidden
  __shared__ float sh_p[NROWS * 128];   // mean_prior
  __shared__ float sh_m[NROWS * 128];   // mean0
  __shared__ float sh_l[NROWS * 128];   // log_std0
  __shared__ float sh_rdeg[NROWS];
  __shared__ float sh_inv[NROWS];
  __shared__ float red[256];

  const int m0 = blockIdx.x * NROWS;
  async_stage(sh_a, h1 + (long long)m0 * 128, NROWS * 128 * 4);
  async_stage(sh_b, z1 + (long long)m0 * 128, NROWS * 128 * 4);
  if (threadIdx.x < NROWS)
    sh_inv[threadIdx.x] = 1.0f / ((float)degree[m0 + threadIdx.x] + 1.0f);
  wait_async0();
  __syncthreads();

  // mean_prior = dm0(z1)
  mlp_layer<128, 128, true>(sh_b, dm0_w1, dm0_b1, sh_t);  __syncthreads();
  mlp_layer<128, 128, false>(sh_t, dm0_w2, dm0_b2, sh_p); __syncthreads();
  // mean0 = rs1(h1)
  mlp_layer<128, 128, true>(sh_a, rs1_w1, rs1_b1, sh_t);  __syncthreads();
  mlp_layer<128, 128, false>(sh_t, rs1_w2, rs1_b2, sh_m); __syncthreads();
  // log_std0 = ds1(h1)
  mlp_layer<128, 128, true>(sh_a, ds1_w1, ds1_b1, sh_t);  __syncthreads();
  mlp_layer<128, 128, false>(sh_t, ds1_w2, ds1_b2, sh_l); __syncthreads();
  // rd1 hidden chain (z1 buffer free now)
  mlp_layer<128, 128, true>(sh_a, rd1_w1, rd1_b1, sh_t);  __syncthreads();
  mlp_layer<128, 128, true>(sh_t, rd1_w2, rd1_b2, sh_b);  __syncthreads();
  if (threadIdx.x < NROWS) {
    float s = rd1_b3[0];
    for (int k = 0; k < 128; ++k) s += sh_b[threadIdx.x * 128 + k] * rd1_w3[k];
    sh_rdeg[threadIdx.x] = fmaxf(s, 0.0f);
  }
  __syncthreads();

  float s2 = 0.0f, k2 = 0.0f;
  for (int i = threadIdx.x; i < NROWS * 128; i += blockDim.x) {
    int m = i >> 7, f = i & 127;
    int node = m0 + m;
    float mp = sh_m[i] + sh_p[i];
    float ls = sh_l[i];
    float nz = noise_normal(1u, (unsigned)(node * 128 + f));
    float zv = mp + nz * __expf(ls);
    float dz = h0[(long long)node * 128 + f] - zv;
    s2 += dz * dz;
    float dm = mp - acc0[(long long)node * 128 + f] * sh_inv[m];
    k2 += -1.0f - 2.0f * ls + dm * dm + __expf(2.0f * ls);
  }
  float S2 = block_reduce(s2, red);
  float K2 = block_reduce(k2, red);
  if (threadIdx.x == 0) {
    atomicAdd(accum + 3, S2);
    atomicAdd(accum + 4, K2);
    float d2 = 0.0f;
    for (int r = 0; r < NROWS; ++r) {
      float dv = sh_rdeg[r] - (float)degree[m0 + r];
      d2 += dv * dv;
    }
    atomicAdd(accum + 5, d2);
  }
}

__global__ void finalize_kernel(const float* __restrict__ accum,
                                float* __restrict__ out, int N) {
  if (threadIdx.x == 0 && blockIdx.x == 0) {
    const float NF = (float)N * 128.0f;
    float loss_self = (accum[0] + accum[3]) / NF;
    float kl = 0.5f * (accum[1] + accum[4]) / NF;
    float loss_deg = (accum[2] + accum[5]) / (float)N;
    out[0] = 0.5f * loss_self + 1e-4f * 0.5f * kl + 10.0f * 0.5f * loss_deg;
  }
}

// ---------------------------------------------------------------------------
// launch
// ---------------------------------------------------------------------------
extern "C" void kernel_launch(void* const* d_in, const int* in_sizes, int n_in,
                              void* d_out, int out_size, void* d_ws, size_t ws_size,
                              hipStream_t stream) {
  const float* h0 = (const float*)d_in[0];
  const int* esrc = (const int*)d_in[1];
  const int* edst = (const int*)d_in[2];
  const int* degree = (const int*)d_in[3];
  const int N = in_sizes[0] / TF;
  const int E = in_sizes[1];

  // params in dict insertion order
  const float* gcn_w0 = (const float*)d_in[4];
  const float* gcn_b0 = (const float*)d_in[5];
  const float* gcn_w1 = (const float*)d_in[6];
  const float* gcn_b1 = (const float*)d_in[7];
  const float* rs0_w1 = (const float*)d_in[8];
  const float* rs0_b1 = (const float*)d_in[9];
  const float* rs0_w2 = (const float*)d_in[10];
  const float* rs0_b2 = (const float*)d_in[11];
  const float* rs1_w1 = (const float*)d_in[12];
  const float* rs1_b1 = (const float*)d_in[13];
  const float* rs1_w2 = (const float*)d_in[14];
  const float* rs1_b2 = (const float*)d_in[15];
  const float* ds0_w1 = (const float*)d_in[16];
  const float* ds0_b1 = (const float*)d_in[17];
  const float* ds0_w2 = (const float*)d_in[18];
  const float* ds0_b2 = (const float*)d_in[19];
  const float* ds1_w1 = (const float*)d_in[20];
  const float* ds1_b1 = (const float*)d_in[21];
  const float* ds1_w2 = (const float*)d_in[22];
  const float* ds1_b2 = (const float*)d_in[23];
  const float* dm0_w1 = (const float*)d_in[24];
  const float* dm0_b1 = (const float*)d_in[25];
  const float* dm0_w2 = (const float*)d_in[26];
  const float* dm0_b2 = (const float*)d_in[27];
  const float* rd0_w1 = (const float*)d_in[28];
  const float* rd0_b1 = (const float*)d_in[29];
  const float* rd0_w2 = (const float*)d_in[30];
  const float* rd0_b2 = (const float*)d_in[31];
  const float* rd0_w3 = (const float*)d_in[32];
  const float* rd0_b3 = (const float*)d_in[33];
  const float* rd1_w1 = (const float*)d_in[34];
  const float* rd1_b1 = (const float*)d_in[35];
  const float* rd1_w2 = (const float*)d_in[36];
  const float* rd1_b2 = (const float*)d_in[37];
  const float* rd1_w3 = (const float*)d_in[38];
  const float* rd1_b3 = (const float*)d_in[39];

  float* ws = (float*)d_ws;
  const long long NF = (long long)N * TF;
  float* accum = ws;            // 16 floats
  float* acc0 = ws + 16;        // [N,128] aggregated h0 (pre-scale)
  float* h1   = acc0 + NF;      // [N,128]
  float* acc1 = h1 + NF;        // [N,128] aggregated h1 (pre-scale)
  float* z1   = acc1 + NF;      // [N,128]
  float* h2w  = z1 + NF;        // [N,64] aligned mirror of h2

  float* out = (float*)d_out;
  float* h2 = out + 1;          // [N,64] (4B-aligned only)

  const int tiles = N / NROWS;                         // 3125
  const long long eth = (long long)E * 32;
  const int eblocks = (int)((eth + 255) / 256);
  const long long n4 = NF / 4;

  zero_accum_kernel<<<1, 32, 0, stream>>>(ws);

  // ---- encoder ----
  copy_f4_kernel<<<2048, 256, 0, stream>>>(h0, acc0, n4);
  agg_edge_kernel<<<eblocks, 256, 0, stream>>>(h0, esrc, edst, acc0, E);
  gcn_gemm_kernel<128, 128, true, false><<<tiles, 256, 0, stream>>>(
      acc0, degree, gcn_w0, gcn_b0, h1, nullptr);
  copy_f4_kernel<<<2048, 256, 0, stream>>>(h1, acc1, n4);
  agg_edge_kernel<<<eblocks, 256, 0, stream>>>(h1, esrc, edst, acc1, E);
  gcn_gemm_kernel<128, 64, false, true><<<tiles, 256, 0, stream>>>(
      acc1, degree, gcn_w1, gcn_b1, h2w, h2);

  // ---- decoder ----
  decoder1_kernel<<<tiles, 256, 0, stream>>>(
      h2w, h1, acc1, degree,
      rs0_w1, rs0_b1, rs0_w2, rs0_b2,
      ds0_w1, ds0_b1, ds0_w2, ds0_b2,
      rd0_w1, rd0_b1, rd0_w2, rd0_b2, rd0_w3, rd0_b3,
      z1, accum);
  decoder2_kernel<<<tiles, 256, 0, stream>>>(
      h0, h1, z1, acc0, degree,
      dm0_w1, dm0_b1, dm0_w2, dm0_b2,
      rs1_w1, rs1_b1, rs1_w2, rs1_b2,
      ds1_w1, ds1_b1, ds1_w2, ds1_b2,
      rd1_w1, rd1_b1, rd1_w2, rd1_b2, rd1_w3, rd1_b3,
      accum);

  finalize_kernel<<<1, 32, 0, stream>>>(accum, out, N);
}